// ScatterReduceSum_57475252355812
// MI455X (gfx1250) — compile-verified
//
#include <hip/hip_runtime.h>

// ScatterReduceSum: out = input; out[index[i,j], j] += src[i,j]
// N_SRC = 1,000,000  N_OUT = 65,536  D = 64
//
// Memory-bound kernel. Strategy:
//  - streamed inputs (src, index) read with non-temporal hint, b128-wide
//  - destination (16 MB) kept L2-resident; updates via no-return
//    global_atomic_add_f32 (fire-and-forget, STOREcnt-tracked)
//  - global_prefetch on the index stream

typedef __attribute__((ext_vector_type(4))) float f32x4;
typedef __attribute__((ext_vector_type(4))) int   i32x4;

__global__ __launch_bounds__(256) void
ScatterReduceSum_copy_kernel(const f32x4* __restrict__ in, f32x4* __restrict__ out, int n4) {
    int t = blockIdx.x * 256 + threadIdx.x;
    if (t < n4) {
        // input is read exactly once -> non-temporal; store regular so the
        // destination lines are resident/dirty in L2 for the atomic phase.
        f32x4 v = __builtin_nontemporal_load(in + t);
        out[t] = v;
    }
}

__global__ __launch_bounds__(256) void
ScatterReduceSum_scatter_kernel(const int* __restrict__ idx,
                                const float* __restrict__ src,
                                float* __restrict__ out,
                                long long n_elems) {
    long long t = (long long)blockIdx.x * 256 + threadIdx.x;
    long long e = t * 4;                       // 4 consecutive elements of one row
    if (e + 3 < n_elems) {
        // prefetch the index stream ~32KB ahead (speculative; dropped if OOB)
        __builtin_prefetch(idx + e + 8192, 0, 0);

        i32x4 i4 = __builtin_nontemporal_load((const i32x4*)(idx + e)); // b128
        f32x4 s4 = __builtin_nontemporal_load((const f32x4*)(src + e)); // b128

        int j = (int)(e & 63);                 // starting column (D = 64, 4 | 64)
        float* base = out + j;

        // out[row*64 + col] += s ; rows are independent per element
        unsafeAtomicAdd(base + ((long long)i4.x << 6) + 0, s4.x);
        unsafeAtomicAdd(base + ((long long)i4.y << 6) + 1, s4.y);
        unsafeAtomicAdd(base + ((long long)i4.z << 6) + 2, s4.z);
        unsafeAtomicAdd(base + ((long long)i4.w << 6) + 3, s4.w);
    }
}

extern "C" void kernel_launch(void* const* d_in, const int* in_sizes, int n_in,
                              void* d_out, int out_size, void* d_ws, size_t ws_size,
                              hipStream_t stream) {
    const float* input = (const float*)d_in[0];   // [N_OUT, D] f32
    const int*   index = (const int*)d_in[1];     // [N_SRC, D] int32 (JAX demotes int64)
    const float* src   = (const float*)d_in[2];   // [N_SRC, D] f32
    float*       out   = (float*)d_out;           // [N_OUT, D] f32

    // Phase 1: out = input  (vectorized f32x4 copy)
    int n_out_elems = in_sizes[0];                // 65536*64 = 4,194,304
    int n4 = n_out_elems / 4;
    int copy_blocks = (n4 + 255) / 256;           // 4096
    ScatterReduceSum_copy_kernel<<<copy_blocks, 256, 0, stream>>>(
        (const f32x4*)input, (f32x4*)out, n4);

    // Phase 2: scatter-add (4 elements per thread)
    long long n_src_elems = (long long)in_sizes[1]; // 1e6*64 = 64,000,000
    long long n_threads = n_src_elems / 4;          // 16,000,000
    int scatter_blocks = (int)((n_threads + 255) / 256); // 62,500
    ScatterReduceSum_scatter_kernel<<<scatter_blocks, 256, 0, stream>>>(
        index, src, out, n_src_elems);
}